// PerfectSpatialHash_41094247088332
// MI455X (gfx1250) — compile-verified
//
#include <hip/hip_runtime.h>

// Two-level perfect spatial hash: pure random-gather, memory-bound kernel.
// Strategy: NT cache policy on streaming data (keep 192MB L2 for the gather
// tables), b128 gathers/stores, 4-way software pipelining per wave32 lane.

typedef __attribute__((ext_vector_type(4))) float v4f;

#define HASH_SIZE   255
#define OFFSET_SIZE 64
#define FEAT        4
#define BLOCK       256
#define PPT         4   // points per thread

__global__ __launch_bounds__(BLOCK)
void PerfectSpatialHash_kernel(const int*   __restrict__ points,            // (N,3)
                               const float* __restrict__ hash_table,        // (255,255,255,4)
                               const int*   __restrict__ offset_table,      // (64,64,64,3)
                               const int*   __restrict__ sparsity_encoding, // (255,255,255)
                               float*       __restrict__ out_feat,          // (N,4)
                               float*       __restrict__ out_sparsity,      // (N,)
                               int n)
{
    const int base = blockIdx.x * (BLOCK * PPT) + threadIdx.x;

    int  px[PPT], py[PPT], pz[PPT];
    bool valid[PPT];
    int  gidx[PPT];

    // ---- Phase 1: streaming point loads (non-temporal: don't pollute L2) ----
#pragma unroll
    for (int k = 0; k < PPT; ++k) {
        int i = base + k * BLOCK;
        valid[k] = (i < n);
        gidx[k]  = i;
        int j = valid[k] ? i : 0;              // safe clamp for tail block
        const int* pp = points + 3 * (size_t)j;
        px[k] = __builtin_nontemporal_load(pp + 0);
        py[k] = __builtin_nontemporal_load(pp + 1);
        pz[k] = __builtin_nontemporal_load(pp + 2);
    }

    // ---- Phase 2: level-1 gather from offset_table (3 MB -> L2 resident) ----
    int ox[PPT], oy[PPT], oz[PPT];
#pragma unroll
    for (int k = 0; k < PPT; ++k) {
        int h1x = px[k] & (OFFSET_SIZE - 1);
        int h1y = py[k] & (OFFSET_SIZE - 1);
        int h1z = pz[k] & (OFFSET_SIZE - 1);
        int oi  = ((h1x * OFFSET_SIZE + h1y) * OFFSET_SIZE + h1z) * 3;
        ox[k] = offset_table[oi + 0];
        oy[k] = offset_table[oi + 1];
        oz[k] = offset_table[oi + 2];
    }

    // ---- Phase 3: level-0 gathers: features (b128) + sparsity encoding ----
    v4f feat[PPT];
    int enc[PPT];
#pragma unroll
    for (int k = 0; k < PPT; ++k) {
        // (p + offset) in [0, 509] -> % 255 via one conditional subtract
        int s0 = px[k] + ox[k]; s0 = (s0 >= HASH_SIZE) ? s0 - HASH_SIZE : s0;
        int s1 = py[k] + oy[k]; s1 = (s1 >= HASH_SIZE) ? s1 - HASH_SIZE : s1;
        int s2 = pz[k] + oz[k]; s2 = (s2 >= HASH_SIZE) ? s2 - HASH_SIZE : s2;
        int hi = (s0 * HASH_SIZE + s1) * HASH_SIZE + s2;
        feat[k] = *(const v4f*)(hash_table + (size_t)hi * FEAT); // global_load_b128
        enc[k]  = sparsity_encoding[hi];
    }

    // ---- Phase 4: sparsity hash (VALU) + NT streaming stores ----
#pragma unroll
    for (int k = 0; k < PPT; ++k) {
        if (!valid[k]) continue;

        // hk = sum_d pf * rsqrt(pf + 1*C1); frac; clip(256*hk, 0, 255) -> int
        const float C1 = 1178101.0f;
        float fx = (float)px[k];
        float fy = (float)py[k];
        float fz = (float)pz[k];
        float hk = fx * rsqrtf(fx + C1)
                 + fy * rsqrtf(fy + C1)
                 + fz * rsqrtf(fz + C1);
        hk = hk - floorf(hk);
        int sh = (int)fminf(fmaxf(256.0f * hk, 0.0f), 255.0f);

        float* fo = out_feat + (size_t)gidx[k] * FEAT;
        __builtin_nontemporal_store(feat[k], (v4f*)fo);          // global_store_b128 NT
        __builtin_nontemporal_store((enc[k] == sh) ? 1.0f : 0.0f,
                                    out_sparsity + gidx[k]);
    }
}

extern "C" void kernel_launch(void* const* d_in, const int* in_sizes, int n_in,
                              void* d_out, int out_size, void* d_ws, size_t ws_size,
                              hipStream_t stream)
{
    const int*   points            = (const int*)  d_in[0];
    const float* hash_table        = (const float*)d_in[1];
    const int*   offset_table      = (const int*)  d_in[2];
    const int*   sparsity_encoding = (const int*)  d_in[3];

    const int n = in_sizes[0] / 3;                 // N_POINTS

    float* out_feat     = (float*)d_out;           // (N,4) features
    float* out_sparsity = out_feat + (size_t)n * FEAT; // (N,) mask as 0/1 float

    const int blocks = (n + BLOCK * PPT - 1) / (BLOCK * PPT);
    PerfectSpatialHash_kernel<<<blocks, BLOCK, 0, stream>>>(
        points, hash_table, offset_table, sparsity_encoding,
        out_feat, out_sparsity, n);
}